// GATLayer_3504693313904
// MI455X (gfx1250) — compile-verified
//
#include <hip/hip_runtime.h>
#include <cstdint>
#include <cstddef>

#define D_MODEL 256
#define HEADS 8
#define NEG_SLOPE 0.2f
#define BN_EPS 1e-5f

typedef __attribute__((ext_vector_type(2))) float v2f;
typedef __attribute__((ext_vector_type(8))) float v8f;

__device__ __forceinline__ float lrelu(float v) { return v > 0.f ? v : NEG_SLOPE * v; }

// Float atomic max via sign-split int/uint atomics (order-preserving bit trick).
__device__ __forceinline__ void atomicMaxF(float* a, float v) {
  if (v >= 0.f) atomicMax((int*)a, __float_as_int(v));
  else          atomicMin((unsigned int*)a, __float_as_uint(v));
}

// ---------------------------------------------------------------------------
// Init: m = -inf, denom = 0, BN accumulators = 0
// ---------------------------------------------------------------------------
__global__ void k_init(float* __restrict__ m, float* __restrict__ denom,
                       float* __restrict__ sum, float* __restrict__ sumsq, int n8) {
  int i = blockIdx.x * 256 + threadIdx.x;
  if (i < n8) {
    m[i] = __int_as_float(0xff800000);  // -inf
    denom[i] = 0.f;
  }
  if (i < D_MODEL) { sum[i] = 0.f; sumsq[i] = 0.f; }
}

// out[n, c] = bias[c]  (accumulation target for the scatter-add)
__global__ void k_bias_init(float* __restrict__ out, const float* __restrict__ bias,
                            long total) {
  long i = (long)blockIdx.x * 256 + threadIdx.x;
  if (i < total) out[i] = bias[i & (D_MODEL - 1)];
}

// ---------------------------------------------------------------------------
// Projection GEMM: xh[N,256] = x[N,256] @ W[256,256], full-f32 WMMA 16x16x4.
// Block: 256 threads = 8 waves. Wave w computes a 16x16 tile at columns
// blockIdx.y*128 + w*16. A tile (16 rows x 256 K) staged once in LDS via the
// gfx1250 async Global->LDS DMA path (ASYNCcnt), bypassing VGPRs.
// ---------------------------------------------------------------------------
__global__ __launch_bounds__(256) void k_gemm(const float* __restrict__ x,
                                              const float* __restrict__ Wm,
                                              float* __restrict__ xh, int N) {
  constexpr int LDA = 260;  // padded stride: 260 % 64 == 4 -> conflict-free column reads
  __shared__ float As[16 * LDA];

  const int row0 = blockIdx.x * 16;
  const int t = threadIdx.x;

  // Stage A asynchronously: thread t covers row (t>>4), 16 floats at col (t&15)*16.
  // INST_OFFSET applies to both the global and LDS addresses (ISA 15.18 async
  // pseudocode), so 4 x b128 with offset 0/16/32/48 stages 64 contiguous bytes.
  {
    int r = t >> 4;
    int cseg = (t & 15) * 16;
    int gr = row0 + r;
    if (gr >= N) gr = N - 1;  // clamp (duplicate loads; guarded on store side)
    unsigned lds = (unsigned)(uintptr_t)(As + r * LDA + cseg);      // 16B aligned
    uint64_t ga = (uint64_t)(uintptr_t)(x + (size_t)gr * D_MODEL + cseg);
    asm volatile(
        "global_load_async_to_lds_b128 %0, %1, off\n\t"
        "global_load_async_to_lds_b128 %0, %1, off offset:16\n\t"
        "global_load_async_to_lds_b128 %0, %1, off offset:32\n\t"
        "global_load_async_to_lds_b128 %0, %1, off offset:48"
        :: "v"(lds), "v"(ga) : "memory");
    asm volatile("s_wait_asynccnt 0" ::: "memory");  // my async writes landed in LDS
  }
  __syncthreads();

  const int wave = t >> 5;
  const int lane = t & 31;
  const int lrow = lane & 15;
  const bool hi = lane >= 16;
  const int col0 = blockIdx.y * 128 + wave * 16;
  const float* Wcol = Wm + col0;

  v8f acc = {};
#pragma unroll 4
  for (int k0 = 0; k0 < D_MODEL; k0 += 4) {
    const int ka = hi ? k0 + 2 : k0;  // lanes 0-15: K0/K1, lanes 16-31: K2/K3
    v2f a = *(const v2f*)(As + lrow * LDA + ka);
    v2f b;
    b.x = Wcol[(size_t)(ka + 0) * D_MODEL + lrow];  // B[k][n], n = lane%16
    b.y = Wcol[(size_t)(ka + 1) * D_MODEL + lrow];
    acc = __builtin_amdgcn_wmma_f32_16x16x4_f32(false, a, false, b, (short)0, acc,
                                                false, false);
  }

  // C/D layout: VGPR i -> M = i (lanes 0-15) / i+8 (lanes 16-31), N = lane%16.
  const int mbase = row0 + (hi ? 8 : 0);
  float* orow = xh + (size_t)mbase * D_MODEL + col0 + lrow;
  if (row0 + 16 <= N) {
    // Full tile: unguarded stores (the common case; N % 16 == 0 here).
#pragma unroll
    for (int i = 0; i < 8; ++i) orow[(size_t)i * D_MODEL] = acc[i];
  } else {
#pragma unroll
    for (int i = 0; i < 8; ++i)
      if (mbase + i < N) orow[(size_t)i * D_MODEL] = acc[i];
  }
}

// ---------------------------------------------------------------------------
// Per-node attention coefficients: one block per node, one wave per head.
// ---------------------------------------------------------------------------
__global__ __launch_bounds__(256) void k_att(const float* __restrict__ xh,
                                             const float* __restrict__ avs,
                                             const float* __restrict__ avd,
                                             float* __restrict__ asrc,
                                             float* __restrict__ adst, int N) {
  int n = blockIdx.x;
  if (n >= N) return;
  int t = threadIdx.x;  // channel 0..255; head = t>>5 (wave32: one head per wave)
  float v = xh[(size_t)n * D_MODEL + t];
  float ps = v * avs[t];
  float pd = v * avd[t];
#pragma unroll
  for (int off = 16; off >= 1; off >>= 1) {
    ps += __shfl_xor(ps, off);
    pd += __shfl_xor(pd, off);
  }
  if ((t & 31) == 0) {
    int h = t >> 5;
    asrc[(size_t)n * HEADS + h] = ps;
    adst[(size_t)n * HEADS + h] = pd;
  }
}

// ---------------------------------------------------------------------------
// Pass 1 over edges (incl. self loops): segment max per destination.
// ---------------------------------------------------------------------------
__global__ __launch_bounds__(256) void k_edge_max(const int* __restrict__ ei,
                                                  const float* __restrict__ asrc,
                                                  const float* __restrict__ adst,
                                                  float* __restrict__ m, int E, int N) {
  long e = (long)blockIdx.x * 256 + threadIdx.x;
  if (e >= (long)E + N) return;
  int s, d;
  if (e < E) { s = ei[e]; d = ei[(long)E + e]; } else { s = d = (int)(e - E); }
#pragma unroll
  for (int h = 0; h < HEADS; ++h) {
    float lg = lrelu(asrc[(size_t)s * HEADS + h] + adst[(size_t)d * HEADS + h]);
    atomicMaxF(&m[(size_t)d * HEADS + h], lg);
  }
}

// Pass 2: exp(logit - max), segment sum.
__global__ __launch_bounds__(256) void k_edge_sum(const int* __restrict__ ei,
                                                  const float* __restrict__ asrc,
                                                  const float* __restrict__ adst,
                                                  const float* __restrict__ m,
                                                  float* __restrict__ denom, int E,
                                                  int N) {
  long e = (long)blockIdx.x * 256 + threadIdx.x;
  if (e >= (long)E + N) return;
  int s, d;
  if (e < E) { s = ei[e]; d = ei[(long)E + e]; } else { s = d = (int)(e - E); }
#pragma unroll
  for (int h = 0; h < HEADS; ++h) {
    float lg = lrelu(asrc[(size_t)s * HEADS + h] + adst[(size_t)d * HEADS + h]);
    atomicAdd(&denom[(size_t)d * HEADS + h], __expf(lg - m[(size_t)d * HEADS + h]));
  }
}

// ---------------------------------------------------------------------------
// Pass 3: weighted message scatter. One wave per edge; lane handles 8 channels
// (lane*8 .. lane*8+7, all inside head lane>>2). xh/out are L2-resident.
// ---------------------------------------------------------------------------
__global__ __launch_bounds__(256) void k_scatter(const int* __restrict__ ei,
                                                 const float* __restrict__ xh,
                                                 const float* __restrict__ asrc,
                                                 const float* __restrict__ adst,
                                                 const float* __restrict__ m,
                                                 const float* __restrict__ denom,
                                                 float* __restrict__ out, int E, int N) {
  int wid = threadIdx.x >> 5;
  int lane = threadIdx.x & 31;
  long e = (long)blockIdx.x * 8 + wid;
  if (e >= (long)E + N) return;
  int s, d;
  if (e < E) { s = ei[e]; d = ei[(long)E + e]; } else { s = d = (int)(e - E); }

  const float* xrow = xh + (size_t)s * D_MODEL;
  __builtin_prefetch(xrow + lane * 8, 0, 1);  // global_prefetch_b8

  int h = lane >> 2;  // 8 channels per lane stay within one 32-wide head
  float lg = lrelu(asrc[(size_t)s * HEADS + h] + adst[(size_t)d * HEADS + h]);
  float alpha = __expf(lg - m[(size_t)d * HEADS + h]) / denom[(size_t)d * HEADS + h];

  int c0 = lane * 8;
  float4 v0 = *(const float4*)(xrow + c0);
  float4 v1 = *(const float4*)(xrow + c0 + 4);
  float* orow = out + (size_t)d * D_MODEL + c0;
  atomicAdd(orow + 0, v0.x * alpha);
  atomicAdd(orow + 1, v0.y * alpha);
  atomicAdd(orow + 2, v0.z * alpha);
  atomicAdd(orow + 3, v0.w * alpha);
  atomicAdd(orow + 4, v1.x * alpha);
  atomicAdd(orow + 5, v1.y * alpha);
  atomicAdd(orow + 6, v1.z * alpha);
  atomicAdd(orow + 7, v1.w * alpha);
}

// ---------------------------------------------------------------------------
// BatchNorm statistics (per-channel sums over nodes) + apply with residual.
// ---------------------------------------------------------------------------
__global__ __launch_bounds__(256) void k_bn_partial(const float* __restrict__ out,
                                                    float* __restrict__ sum,
                                                    float* __restrict__ sumsq, int N) {
  int c = threadIdx.x;
  int r0 = blockIdx.x * 256;
  int r1 = r0 + 256;
  if (r1 > N) r1 = N;
  float s = 0.f, ss = 0.f;
  for (int r = r0; r < r1; ++r) {
    float v = out[(size_t)r * D_MODEL + c];
    s += v;
    ss += v * v;
  }
  atomicAdd(sum + c, s);
  atomicAdd(sumsq + c, ss);
}

__global__ __launch_bounds__(256) void k_bn_apply(float* __restrict__ out,
                                                  const float* __restrict__ x,
                                                  const float* __restrict__ sum,
                                                  const float* __restrict__ sumsq,
                                                  const float* __restrict__ gamma,
                                                  const float* __restrict__ beta, int N) {
  long i = (long)blockIdx.x * 256 + threadIdx.x;
  if (i >= (long)N * D_MODEL) return;
  int c = (int)(i & (D_MODEL - 1));
  float invN = 1.0f / (float)N;
  float mean = sum[c] * invN;
  float var = sumsq[c] * invN - mean * mean;
  float v = (out[i] - mean) * rsqrtf(var + BN_EPS) * gamma[c] + beta[c] + x[i];
  out[i] = v;
}

// ---------------------------------------------------------------------------
extern "C" void kernel_launch(void* const* d_in, const int* in_sizes, int n_in,
                              void* d_out, int out_size, void* d_ws, size_t ws_size,
                              hipStream_t stream) {
  const float* x       = (const float*)d_in[0];
  const int*   ei      = (const int*)d_in[1];
  const float* W       = (const float*)d_in[2];
  const float* att_src = (const float*)d_in[3];
  const float* att_dst = (const float*)d_in[4];
  const float* bias    = (const float*)d_in[5];
  const float* gamma   = (const float*)d_in[6];
  const float* beta    = (const float*)d_in[7];
  float* out = (float*)d_out;

  const int N = in_sizes[0] / D_MODEL;
  const int E = in_sizes[1] / 2;
  const long EN = (long)E + N;

  // Workspace carve-out.
  char* w = (char*)d_ws;
  float* xh    = (float*)w; w += (size_t)N * D_MODEL * sizeof(float);
  float* asrc  = (float*)w; w += (size_t)N * HEADS * sizeof(float);
  float* adst  = (float*)w; w += (size_t)N * HEADS * sizeof(float);
  float* mmax  = (float*)w; w += (size_t)N * HEADS * sizeof(float);
  float* denom = (float*)w; w += (size_t)N * HEADS * sizeof(float);
  float* bsum  = (float*)w; w += D_MODEL * sizeof(float);
  float* bsq   = (float*)w; w += D_MODEL * sizeof(float);

  const int n8 = N * HEADS;
  k_init<<<(n8 + 255) / 256, 256, 0, stream>>>(mmax, denom, bsum, bsq, n8);

  const long total = (long)N * D_MODEL;
  k_bias_init<<<(int)((total + 255) / 256), 256, 0, stream>>>(out, bias, total);

  dim3 ggrid((N + 15) / 16, D_MODEL / 128);
  k_gemm<<<ggrid, 256, 0, stream>>>(x, W, xh, N);

  k_att<<<N, 256, 0, stream>>>(xh, att_src, att_dst, asrc, adst, N);

  k_edge_max<<<(int)((EN + 255) / 256), 256, 0, stream>>>(ei, asrc, adst, mmax, E, N);
  k_edge_sum<<<(int)((EN + 255) / 256), 256, 0, stream>>>(ei, asrc, adst, mmax, denom, E, N);

  k_scatter<<<(int)((EN + 7) / 8), 256, 0, stream>>>(ei, xh, asrc, adst, mmax, denom,
                                                     out, E, N);

  k_bn_partial<<<(N + 255) / 256, 256, 0, stream>>>(out, bsum, bsq, N);
  k_bn_apply<<<(int)((total + 255) / 256), 256, 0, stream>>>(out, x, bsum, bsq, gamma,
                                                             beta, N);
}